// CNNModel_76312978915482
// MI455X (gfx1250) — compile-verified
//
#include <hip/hip_runtime.h>
#include <stdint.h>

// Fused anomaly-discord kernel for MI455X (gfx1250).
// out[b,r,c] = cond[b,(r+1)>>2,(c+1)>>2] ? 0 : 1, where cond is computed from
// the disjoint 4x4 input window of each pooled cell (see analysis).
// Bandwidth-bound: ~212 MB total traffic -> ~9us roofline at 23.3 TB/s.
// Uses CDNA5 async global->LDS copies (ASYNCcnt) to stage each 4x314 strip.

namespace {
constexpr int Bsz = 256;   // batch
constexpr int H   = 330;
constexpr int W   = 314;
constexpr int HP  = 83;    // pooled rows
constexpr int WP  = 79;    // pooled cols
constexpr int NT  = 256;   // threads per block (8 waves of 32)
}

typedef float v2f __attribute__((ext_vector_type(2)));  // native vector for NT store

__device__ __forceinline__ void async_copy_b64(unsigned lds_byte_addr, const void* gaddr) {
  // CDNA5 async load: copies 8 bytes from the global address in vaddr to the LDS address in vdst, tracked by ASYNCcnt.
  asm volatile("global_load_async_to_lds_b64 %0, %1, off"
               :: "v"(lds_byte_addr), "v"(gaddr)
               : "memory");
}

__device__ __forceinline__ void wait_async0() {
#if __has_builtin(__builtin_amdgcn_s_wait_asynccnt)
  __builtin_amdgcn_s_wait_asynccnt(0);
#else
  asm volatile("s_wait_asynccnt 0" ::: "memory");
#endif
}

__global__ __launch_bounds__(NT) void anomaly_fused(const float* __restrict__ x,
                                                    const float* __restrict__ lb,
                                                    const float* __restrict__ q1,
                                                    float* __restrict__ out) {
  __shared__ float tile[4 * W];       // window rows 4i-1..4i+2 (cols 0..313)
  __shared__ unsigned condS[WP];

  const int i = blockIdx.x;           // pooled row
  const int b = blockIdx.y;           // batch
  const int t = threadIdx.x;

  const int r0    = 4 * i - 1;                      // first window row (may be -1)
  const int first = (r0 < 0) ? 0 : r0;              // first in-range x/out row
  const int last  = (r0 + 3 > H - 1) ? (H - 1) : (r0 + 3);
  const int nflt  = (last - first + 1) * W;         // 942 (edges) or 1256 floats
  const int ldsStart = (first - r0) * W;            // 0 or 314 (float index)

  // Zero-pad LDS rows corresponding to conv's zero padding (row -1 / row H).
  if (r0 < 0)
    for (int k = t; k < W; k += NT) tile[k] = 0.0f;
  if (r0 + 3 >= H)
    for (int k = t; k < W; k += NT) tile[3 * W + k] = 0.0f;

  // Async-copy the contiguous in-range strip (rows are consecutive in memory).
  // Base is always 8B-aligned: row pitch = 314*4 = 1256 B (== 0 mod 8).
  const float* gbase = x + ((size_t)b * H + first) * (size_t)W;
  const int n64 = nflt >> 1;                        // 471 or 628 b64 transfers
  for (int k = t; k < n64; k += NT) {
    unsigned laddr = (unsigned)(size_t)&tile[ldsStart + 2 * k];  // low 32 bits are the LDS address
    async_copy_b64(laddr, gbase + 2 * k);
  }
  wait_async0();
  __syncthreads();

  // One pooled cell per lane (j = 0..78).
  if (t < WP) {
    const int j = t;
    float w[4][4];
#pragma unroll
    for (int rr = 0; rr < 4; ++rr) {
#pragma unroll
      for (int cc = 0; cc < 4; ++cc) {
        const int col = 4 * j - 1 + cc;             // -1..314; OOB -> conv zero-pad
        w[rr][cc] = (col >= 0 && col < W) ? tile[rr * W + col] : 0.0f;
      }
    }
    // Four 2x2 conv sums of this cell's pooling window.
    const float s00 = (w[0][0] + w[0][1]) + (w[1][0] + w[1][1]);
    const float s01 = (w[0][2] + w[0][3]) + (w[1][2] + w[1][3]);
    const float s10 = (w[2][0] + w[2][1]) + (w[3][0] + w[3][1]);
    const float s11 = (w[2][2] + w[2][3]) + (w[3][2] + w[3][3]);
    const float mean = 0.25f * ((s00 + s01) + (s10 + s11));
    // max-pool of |min(conv,0)| = max(0, max(-s_k))
    const float m   = fmaxf(fmaxf(fmaxf(-s00, -s01), fmaxf(-s10, -s11)), 0.0f);
    const float neg = -m;
    const float lbv = lb[i * WP + j];
    const float qv  = q1[i * WP + j];
    const bool cond = (neg < lbv) && ((mean / neg) > (qv / lbv));
    condS[j] = cond ? 1u : 0u;
  }
  __syncthreads();

  // Streamed, nontemporal output: rows first..last map to this cell-row i.
  float* obase = out + ((size_t)b * H + first) * (size_t)W;
  for (int k = t; k < n64; k += NT) {
    const int c0 = (2 * k) % W;                     // W even -> v2f never splits rows
    v2f v;
    v.x = condS[(c0 + 1) >> 2] ? 0.0f : 1.0f;
    v.y = condS[(c0 + 2) >> 2] ? 0.0f : 1.0f;
    __builtin_nontemporal_store(v, reinterpret_cast<v2f*>(obase + 2 * k));
  }
}

extern "C" void kernel_launch(void* const* d_in, const int* in_sizes, int n_in,
                              void* d_out, int out_size, void* d_ws, size_t ws_size,
                              hipStream_t stream) {
  // Inputs (setup_inputs order): x, lower_bound1, q1, map_rows, map_cols.
  // map_rows/map_cols are deterministic (derived analytically in-kernel).
  const float* x  = (const float*)d_in[0];
  const float* lb = (const float*)d_in[1];
  const float* q1 = (const float*)d_in[2];
  (void)in_sizes; (void)n_in; (void)d_ws; (void)ws_size; (void)out_size;

  dim3 grid(HP, Bsz);   // one block per (pooled-row, batch)
  anomaly_fused<<<grid, NT, 0, stream>>>(x, lb, q1, (float*)d_out);
}